// GCNMolGCN_48962627175096
// MI455X (gfx1250) — compile-verified
//
#include <hip/hip_runtime.h>
#include <hip/hip_bf16.h>

typedef __attribute__((ext_vector_type(2))) float v2f;
typedef __attribute__((ext_vector_type(8))) float v8f;

// ---------------------------------------------------------------------------
// Degree / normalization: deg includes self-loop (init 1.0), dinv = rsqrt(deg)
// ---------------------------------------------------------------------------
__global__ void k_deg_init(float* __restrict__ dinv, int N) {
    int i = blockIdx.x * blockDim.x + threadIdx.x;
    if (i < N) dinv[i] = 1.0f;  // self-loop contribution
}

__global__ void k_deg_count(const int* __restrict__ dst, float* __restrict__ dinv, int E) {
    int e = blockIdx.x * blockDim.x + threadIdx.x;
    if (e < E) atomicAdd(&dinv[dst[e]], 1.0f);
}

__global__ void k_deg_rsqrt(float* __restrict__ dinv, int N) {
    int i = blockIdx.x * blockDim.x + threadIdx.x;
    if (i < N) dinv[i] = rsqrtf(dinv[i]);  // deg >= 1 always (self-loop)
}

// ---------------------------------------------------------------------------
// Dense transform T = act(A + bias) (NxK) @ B (KxM), fp32 WMMA 16x16x4.
// One wave per 16-row strip; it computes ALL M/16 column tiles so each A
// fragment is loaded exactly once (register-blocked: M/16 x 8 acc VGPRs).
// HAS_BIAS fuses the PREVIOUS layer's epilogue: a = relu(a + bias[k]),
// hoisted to a template so the K-loop is branch-free.
// A frag: lane&15 = M row, (lane>>4)*2 = K pair; B frag mirrored.
// Strip guard is wave-uniform => EXEC all-ones inside WMMA (ISA requirement).
// ---------------------------------------------------------------------------
template <int K, int M, bool HAS_BIAS>
__global__ void k_gemm_wmma_f32(const float* __restrict__ A, const float* __restrict__ B,
                                float* __restrict__ D, int N,
                                const float* __restrict__ bias) {
    constexpr int TM = M / 16;
    const int lane  = threadIdx.x & 31;
    const int wave  = threadIdx.x >> 5;
    const int strip = blockIdx.x * (blockDim.x >> 5) + wave;
    if (strip >= (N >> 4)) return;           // uniform per wave

    const int l15 = lane & 15;
    const int kh  = (lane >> 4) << 1;        // 0 or 2
    const float* arow = A + (size_t)(strip * 16 + l15) * K + kh;  // A[m][kh + k]
    const float* brow = B + (size_t)kh * M + l15;                 // B[kh + k][col]

    v8f acc[TM];
#pragma unroll
    for (int j = 0; j < TM; ++j) acc[j] = (v8f){};

#pragma unroll 4
    for (int k = 0; k < K; k += 4) {
        float a0 = arow[k];
        float a1 = arow[k + 1];
        if constexpr (HAS_BIAS) {            // fused bias+ReLU of previous layer
            a0 = fmaxf(a0 + bias[k + kh], 0.0f);
            a1 = fmaxf(a1 + bias[k + kh + 1], 0.0f);
        }
        v2f a; a.x = a0; a.y = a1;
#pragma unroll
        for (int j = 0; j < TM; ++j) {
            v2f b;
            b.x = brow[k * M + j * 16];
            b.y = brow[k * M + M + j * 16];
            acc[j] = __builtin_amdgcn_wmma_f32_16x16x4_f32(
                /*neg_a=*/false, a, /*neg_b=*/false, b,
                /*c_mod=*/(short)0, acc[j], /*reuse_a=*/false, /*reuse_b=*/false);
        }
    }

    // C/D layout: VGPR r -> (M=r, lanes 0-15), (M=r+8, lanes 16-31)
    const int rbase = strip * 16 + ((lane >> 4) << 3);
#pragma unroll
    for (int j = 0; j < TM; ++j) {
        float* drow = D + (size_t)rbase * M + j * 16 + l15;
#pragma unroll
        for (int r = 0; r < 8; ++r) drow[(size_t)r * M] = acc[j][r];
    }
}

// ---------------------------------------------------------------------------
// Aggregation: H[i] = T[i]*dinv[i]^2 (self loop), then scatter-add edges.
// ---------------------------------------------------------------------------
__global__ void k_agg_self(const float* __restrict__ T, const float* __restrict__ dinv,
                           float* __restrict__ H, int N, int C) {
    int idx = blockIdx.x * blockDim.x + threadIdx.x;
    if (idx >= N * C) return;
    float d = dinv[idx / C];
    H[idx] = T[idx] * d * d;
}

// One thread per (edge, 4-channel group). Consecutive threads cover one edge's
// row => coalesced 512B gathers from T[src]; 4 fp32 atomics into H[dst].
// T and H (51.2 MB each) are L2-resident, so this phase runs near L2 speed.
__global__ void k_agg_edges(const float* __restrict__ T, const int* __restrict__ src,
                            const int* __restrict__ dst, const float* __restrict__ dinv,
                            float* __restrict__ H, int E, int C) {
    const int groups = C >> 2;
    int tid = blockIdx.x * blockDim.x + threadIdx.x;
    int e = tid / groups;
    int g = tid - e * groups;
    if (e >= E) return;
    int s = src[e], d = dst[e];
    float norm = dinv[s] * dinv[d];
    const float4 v = ((const float4*)(T + (size_t)s * C))[g];
    float* hp = H + (size_t)d * C + (g << 2);
    atomicAdd(hp + 0, v.x * norm);
    atomicAdd(hp + 1, v.y * norm);
    atomicAdd(hp + 2, v.z * norm);
    atomicAdd(hp + 3, v.w * norm);
}

// ---------------------------------------------------------------------------
// Min-pool over nodes (N x 64 -> 64) with fused final bias, two passes.
// ---------------------------------------------------------------------------
__global__ void k_minpool_part(const float* __restrict__ H, const float* __restrict__ b3,
                               float* __restrict__ part, int N) {
    __shared__ float sm[256];
    const int t = threadIdx.x;
    const int c = t & 63;
    const int rg = t >> 6;                   // 4 rows per block-iteration
    const float bc = b3[c];
    float m = __builtin_inff();
    for (int i = blockIdx.x * 4 + rg; i < N; i += gridDim.x * 4)
        m = fminf(m, H[(size_t)i * 64 + c] + bc);
    sm[t] = m;
    __syncthreads();
    if (t < 128) sm[t] = fminf(sm[t], sm[t + 128]);
    __syncthreads();
    if (t < 64) part[blockIdx.x * 64 + c] = fminf(sm[t], sm[t + 64]);
}

__global__ void k_minpool_final(const float* __restrict__ part, float* __restrict__ out, int G) {
    int c = threadIdx.x;                     // 64 threads
    float m = __builtin_inff();
    for (int g = 0; g < G; ++g) m = fminf(m, part[g * 64 + c]);
    out[c] = m;
}

// ---------------------------------------------------------------------------
extern "C" void kernel_launch(void* const* d_in, const int* in_sizes, int n_in,
                              void* d_out, int out_size, void* d_ws, size_t ws_size,
                              hipStream_t stream) {
    const float* x  = (const float*)d_in[0];
    const int*  ei  = (const int*)d_in[1];
    const float* W1 = (const float*)d_in[2];
    const float* b1 = (const float*)d_in[3];
    const float* W2 = (const float*)d_in[4];
    const float* b2 = (const float*)d_in[5];
    const float* W3 = (const float*)d_in[6];
    const float* b3 = (const float*)d_in[7];
    float* out = (float*)d_out;

    const int IN = 128;
    const int N = in_sizes[0] / IN;          // 100000 (divisible by 16)
    const int E = in_sizes[1] / 2;           // 1600000
    const int* src = ei;
    const int* dst = ei + E;

    // Workspace layout (fp32): dinv[N] | T[N*128] | H[N*128] | part[256*64]
    float* dinv = (float*)d_ws;
    size_t npad = ((size_t)N + 3) & ~(size_t)3;   // keep 16B alignment for T
    float* T    = dinv + npad;
    float* H    = T + (size_t)N * 128;
    float* part = H + (size_t)N * 128;

    const int TB = 256;                      // 8 waves per block
    const int WPB = TB / 32;
    const int strips = N / 16;
    const int gemmBlocks = (strips + WPB - 1) / WPB;

    // --- degree / normalization ---
    k_deg_init <<<(N + TB - 1) / TB, TB, 0, stream>>>(dinv, N);
    k_deg_count<<<(E + TB - 1) / TB, TB, 0, stream>>>(dst, dinv, E);
    k_deg_rsqrt<<<(N + TB - 1) / TB, TB, 0, stream>>>(dinv, N);

    auto aggregate = [&](int M) {
        int nc = N * M;
        k_agg_self<<<(nc + TB - 1) / TB, TB, 0, stream>>>(T, dinv, H, N, M);
        long ethreads = (long)E * (M / 4);
        k_agg_edges<<<(int)((ethreads + TB - 1) / TB), TB, 0, stream>>>(
            T, src, dst, dinv, H, E, M);
    };

    // Layer 1: T = x @ W1 ; H = aggregate(T)
    k_gemm_wmma_f32<128, 128, false><<<gemmBlocks, TB, 0, stream>>>(x, W1, T, N, nullptr);
    aggregate(128);
    // Layer 2: T = relu(H + b1) @ W2 ; H = aggregate(T)
    k_gemm_wmma_f32<128, 128, true><<<gemmBlocks, TB, 0, stream>>>(H, W2, T, N, b1);
    aggregate(128);
    // Layer 3: T = relu(H + b2) @ W3 ; H = aggregate(T)
    k_gemm_wmma_f32<128, 64, true><<<gemmBlocks, TB, 0, stream>>>(H, W3, T, N, b2);
    aggregate(64);

    // out = min over nodes of (H + b3)
    k_minpool_part <<<256, 256, 0, stream>>>(H, b3, part, N);
    k_minpool_final<<<1, 64, 0, stream>>>(part, out, 256);
}